// SO2_Linear_44246753084081
// MI455X (gfx1250) — compile-verified
//
#include <hip/hip_runtime.h>
#include <hip/hip_bf16.h>

typedef float v2f __attribute__((ext_vector_type(2)));
typedef float v8f __attribute__((ext_vector_type(8)));

#define FEAT 480
#define ROWS 16
#define NTHREADS 256

// LDS layout (floats). Strides 228/196/68 = 4*odd (mod 64) -> 16 rows hit 16
// distinct bank-pairs, and row bases are 8B-aligned so A pairs load as b64.
#define LD0 228
#define LD1 196
#define LD2 68
#define OFF_A0 0
#define OFF_A1 (OFF_A0 + 16*LD0)       // 3648
#define OFF_A2 (OFF_A1 + 16*LD1)       // 6784
#define OFF_Y0 (OFF_A2 + 16*LD2)       // 7872
#define OFF_Y1 (OFF_Y0 + 16*LD0)       // 11520
#define OFF_Y2 (OFF_Y1 + 16*LD1)       // 14656
#define OFF_R1 (OFF_Y2 + 16*LD2)       // 15744
#define OFF_R2 (OFF_R1 + 16*9)         // 15888
#define SMEM_FLOATS (OFF_R2 + 16*25)   // 16288 floats = 65152 B
#define OFF_XS OFF_Y0                  // staged raw x tile (7680 <= 7872 floats)

template<int D>
__device__ inline void mm(const float* A, const float* B, float* C) {
#pragma unroll
  for (int i = 0; i < D; ++i)
#pragma unroll
    for (int j = 0; j < D; ++j) {
      float s = 0.f;
#pragma unroll
      for (int k = 0; k < D; ++k) s += A[i*D + k] * B[k*D + j];
      C[i*D + j] = s;
    }
}

template<int L>
__device__ inline void zrot(float ang, float* M) {
  constexpr int D = 2*L + 1;
#pragma unroll
  for (int i = 0; i < D*D; ++i) M[i] = 0.f;
#pragma unroll
  for (int i = 0; i < D; ++i) {
    float f = (float)(L - i);
    M[i*D + i] = cosf(f * ang);
    int rv = D - 1 - i;
    if (rv != i) M[i*D + rv] = sinf(f * ang);
  }
}

// rot = Zrot(alpha) @ J @ Zrot(beta) @ J  (gamma = 0 -> identity)
template<int L>
__device__ inline void wigner(float a, float b, const float* J, float* rot) {
  constexpr int D = 2*L + 1;
  float za[D*D], zb[D*D], t1[D*D];
  zrot<L>(a, za);
  zrot<L>(b, zb);
  mm<D>(za, J, t1);
  mm<D>(t1, zb, za);
  mm<D>(za, J, rot);
}

// ---------------------------------------------------------------------------
// Weight prep: pack each GEMM's B matrix into WMMA-ready pair order.
// For K-block kb=4*k4 and lane-half hi, lane (col o) needs B[kb+hi][o] and
// B[kb+hi+2][o]; store them adjacent:
//   Bp[ ((k4*2 + hi)*NO + o)*2 + j ] = W[o][k4*4 + hi + 2*j]
// so the GEMM loop does a single global_load_b64 per WMMA.
// ---------------------------------------------------------------------------
__global__ void prep_weights(const float* __restrict__ w0, const float* __restrict__ w1,
                             const float* __restrict__ w2, float* __restrict__ ws) {
  float* Bp0 = ws;
  float* Bp1 = ws + 224*224;
  float* Bp2 = Bp1 + 192*192;
  const int total = 224*224 + 192*192 + 64*64;
  for (int id = blockIdx.x*blockDim.x + threadIdx.x; id < total; id += gridDim.x*blockDim.x) {
    if (id < 224*224) {
      const int NO = 224;
      int j = id & 1, p = id >> 1;
      int o = p % NO, t = p / NO, hi = t & 1, k4 = t >> 1;
      int k = k4*4 + hi + 2*j;
      Bp0[id] = w0[o*224 + k];
    } else if (id < 224*224 + 192*192) {
      const int NO = 192;
      int jj = id - 224*224;
      int j = jj & 1, p = jj >> 1;
      int o = p % NO, t = p / NO, hi = t & 1, k4 = t >> 1;
      int k = k4*4 + hi + 2*j;
      // Wc1 = [[Wr, -Wi], [Wi, Wr]] from w1 (192 x 96)
      float v;
      if (o < 96) v = (k < 96) ? w1[o*96 + k] : -w1[(96 + o)*96 + (k - 96)];
      else        v = (k < 96) ? w1[o*96 + k] :  w1[(o - 96)*96 + (k - 96)];
      Bp1[jj] = v;
    } else {
      const int NO = 64;
      int jj = id - 224*224 - 192*192;
      int j = jj & 1, p = jj >> 1;
      int o = p % NO, t = p / NO, hi = t & 1, k4 = t >> 1;
      int k = k4*4 + hi + 2*j;
      float v;
      if (o < 32) v = (k < 32) ? w2[o*32 + k] : -w2[(32 + o)*32 + (k - 32)];
      else        v = (k < 32) ? w2[o*32 + k] :  w2[(o - 32)*32 + (k - 32)];
      Bp2[jj] = v;
    }
  }
}

// One 16x16 output tile: D = A(16xK) * B(Kx16) [+ bias], f32 WMMA, K%4==0.
template<int K, int NO, int LDA>
__device__ inline void gemm_tile(const float* __restrict__ sA, float* __restrict__ sY,
                                 const float* __restrict__ Bp, const float* __restrict__ bias,
                                 int o0, int row, int hi) {
  v8f acc = {0.f, 0.f, 0.f, 0.f, 0.f, 0.f, 0.f, 0.f};
  const float* arow = sA + row*LDA;
  const v2f* B2 = (const v2f*)Bp;
  const int bcol = o0 + row;
  // speculative prefetch of this tile's B stream (uniform, no branch)
  __builtin_prefetch(&Bp[2*bcol], 0, 3);
#pragma unroll 4
  for (int k4 = 0; k4 < K/4; ++k4) {
    v2f a = *(const v2f*)(arow + k4*4 + 2*hi);       // ds_load_b64
    v2f b = B2[(k4*2 + hi)*NO + bcol];               // global_load_b64
    acc = __builtin_amdgcn_wmma_f32_16x16x4_f32(
        false, a, false, b, (short)0, acc, false, false);
  }
  float bv = bias ? bias[o0 + row] : 0.f;
#pragma unroll
  for (int r = 0; r < 8; ++r)
    sY[(r + 8*hi)*LDA + o0 + row] = acc[r] + bv;     // D: M = r + 8*hi, N = row
}

__global__ __launch_bounds__(NTHREADS)
void so2_fused(const float* __restrict__ x, const float* __restrict__ R,
               const float* __restrict__ Jf, const float* __restrict__ b0,
               const float* __restrict__ Bp0, const float* __restrict__ Bp1,
               const float* __restrict__ Bp2, float* __restrict__ out, int N) {
  __shared__ float sm[SMEM_FLOATS];
  const int tid = threadIdx.x;
  const int n0 = blockIdx.x * ROWS;

  // ---- Phase 0a: stage x tile into LDS (coalesced) ----
  const long long base = (long long)n0 * FEAT;
  const long long lim  = (long long)N * FEAT;
  for (int i = tid; i < ROWS*FEAT; i += NTHREADS) {
    long long g = base + i;
    sm[OFF_XS + i] = (g < lim) ? x[g] : 0.f;
  }

  // ---- Phase 0b: per-sample Wigner rotations (threads 0..15) ----
  if (tid < ROWS) {
    int n = n0 + tid; if (n >= N) n = N - 1;
    float rx = R[3*n + 0], ry = R[3*n + 1], rz = R[3*n + 2];
    float inv = rsqrtf(rx*rx + ry*ry + rz*rz);
    float v0 = fminf(fmaxf(ry*inv, -1.f), 1.f);   // v = R[:, [1,2,0]] normalized
    float v1 = fminf(fmaxf(rz*inv, -1.f), 1.f);
    float v2 = fminf(fmaxf(rx*inv, -1.f), 1.f);
    float alpha = atan2f(v0, v2);
    float beta  = acosf(v1);

    float J1[9], J2[25], rot1[9], rot2[25];
#pragma unroll
    for (int i = 0; i < 3; ++i)
#pragma unroll
      for (int j = 0; j < 3; ++j) J1[i*3 + j] = Jf[(1 + i)*9 + (1 + j)];
#pragma unroll
    for (int i = 0; i < 5; ++i)
#pragma unroll
      for (int j = 0; j < 5; ++j) J2[i*5 + j] = Jf[(4 + i)*9 + (4 + j)];

    wigner<1>(alpha, beta, J1, rot1);
    wigner<2>(alpha, beta, J2, rot2);
#pragma unroll
    for (int k = 0; k < 9; ++k)  sm[OFF_R1 + tid*9  + k] = rot1[k];
#pragma unroll
    for (int k = 0; k < 25; ++k) sm[OFF_R2 + tid*25 + k] = rot2[k];
  }
  __syncthreads();

  // ---- Phase 1: forward-rotate + pack into m-gathered A tiles ----
  for (int i = tid; i < 16*128; i += NTHREADS) {        // l=0 passthrough
    int s = i >> 7, k = i & 127;
    sm[OFF_A0 + s*LD0 + k] = sm[OFF_XS + s*FEAT + k];
  }
  for (int i = tid; i < 16*64; i += NTHREADS) {         // l=1: blk @ rot1
    int s = i >> 6, c = i & 63;
    const float* xb  = &sm[OFF_XS + s*FEAT + 128 + 3*c];
    const float* rot = &sm[OFF_R1 + s*9];
    float y0 = 0.f, y1 = 0.f, y2 = 0.f;
#pragma unroll
    for (int d = 0; d < 3; ++d) {
      float xv = xb[d];
      y0 += xv * rot[d*3 + 0];
      y1 += xv * rot[d*3 + 1];
      y2 += xv * rot[d*3 + 2];
    }
    sm[OFF_A1 + s*LD1 + c]       = y0;  // m=1 real
    sm[OFF_A0 + s*LD0 + 128 + c] = y1;  // m=0 center
    sm[OFF_A1 + s*LD1 + 96 + c]  = y2;  // m=1 imag
  }
  for (int i = tid; i < 16*32; i += NTHREADS) {         // l=2: blk @ rot2
    int s = i >> 5, c = i & 31;
    const float* xb  = &sm[OFF_XS + s*FEAT + 320 + 5*c];
    const float* rot = &sm[OFF_R2 + s*25];
    float y[5] = {0.f, 0.f, 0.f, 0.f, 0.f};
#pragma unroll
    for (int d = 0; d < 5; ++d) {
      float xv = xb[d];
#pragma unroll
      for (int e = 0; e < 5; ++e) y[e] += xv * rot[d*5 + e];
    }
    sm[OFF_A2 + s*LD2 + c]       = y[0]; // m=2 real
    sm[OFF_A1 + s*LD1 + 64 + c]  = y[1]; // m=1 real (l=2)
    sm[OFF_A0 + s*LD0 + 192 + c] = y[2]; // m=0 center
    sm[OFF_A1 + s*LD1 + 160 + c] = y[3]; // m=1 imag (l=2)
    sm[OFF_A2 + s*LD2 + 32 + c]  = y[4]; // m=2 imag
  }
  __syncthreads();

  // ---- Phase 2: 30 WMMA col-tiles over 8 waves, specialized per class ----
  const int wave = tid >> 5, lane = tid & 31;
  const int row = lane & 15, hi = lane >> 4;
  for (int t = wave; t < 30; t += 8) {
    if (t < 14)
      gemm_tile<224, 224, LD0>(&sm[OFF_A0], &sm[OFF_Y0], Bp0, b0, t*16, row, hi);
    else if (t < 26)
      gemm_tile<192, 192, LD1>(&sm[OFF_A1], &sm[OFF_Y1], Bp1, nullptr, (t - 14)*16, row, hi);
    else
      gemm_tile<64, 64, LD2>(&sm[OFF_A2], &sm[OFF_Y2], Bp2, nullptr, (t - 26)*16, row, hi);
  }
  __syncthreads();

  // ---- Phase 3: inverse rotate (blk @ rot^T) + scatter to out ----
  for (int i = tid; i < 16*128; i += NTHREADS) {        // l=0
    int s = i >> 7, k = i & 127;
    int n = n0 + s;
    if (n < N) out[(long long)n*FEAT + k] = sm[OFF_Y0 + s*LD0 + k];
  }
  for (int i = tid; i < 16*64; i += NTHREADS) {         // l=1
    int s = i >> 6, c = i & 63;
    int n = n0 + s;
    if (n >= N) continue;
    const float* rot = &sm[OFF_R1 + s*9];
    float in0 = sm[OFF_Y1 + s*LD1 + c];
    float in1 = sm[OFF_Y0 + s*LD0 + 128 + c];
    float in2 = sm[OFF_Y1 + s*LD1 + 96 + c];
    float* op = &out[(long long)n*FEAT + 128 + 3*c];
#pragma unroll
    for (int e = 0; e < 3; ++e)
      op[e] = in0*rot[e*3 + 0] + in1*rot[e*3 + 1] + in2*rot[e*3 + 2];
  }
  for (int i = tid; i < 16*32; i += NTHREADS) {         // l=2
    int s = i >> 5, c = i & 31;
    int n = n0 + s;
    if (n >= N) continue;
    const float* rot = &sm[OFF_R2 + s*25];
    float in[5];
    in[0] = sm[OFF_Y2 + s*LD2 + c];
    in[1] = sm[OFF_Y1 + s*LD1 + 64 + c];
    in[2] = sm[OFF_Y0 + s*LD0 + 192 + c];
    in[3] = sm[OFF_Y1 + s*LD1 + 160 + c];
    in[4] = sm[OFF_Y2 + s*LD2 + 32 + c];
    float* op = &out[(long long)n*FEAT + 320 + 5*c];
#pragma unroll
    for (int e = 0; e < 5; ++e) {
      float s5 = 0.f;
#pragma unroll
      for (int d = 0; d < 5; ++d) s5 += in[d] * rot[e*5 + d];
      op[e] = s5;
    }
  }
}

extern "C" void kernel_launch(void* const* d_in, const int* in_sizes, int n_in,
                              void* d_out, int out_size, void* d_ws, size_t ws_size,
                              hipStream_t stream) {
  const float* x  = (const float*)d_in[0];
  const float* R  = (const float*)d_in[1];
  const float* Jf = (const float*)d_in[2];
  const float* w0 = (const float*)d_in[3];
  const float* b0 = (const float*)d_in[4];
  const float* w1 = (const float*)d_in[5];
  const float* w2 = (const float*)d_in[6];
  float* out = (float*)d_out;
  float* ws  = (float*)d_ws;
  const int N = in_sizes[0] / FEAT;

  prep_weights<<<120, 256, 0, stream>>>(w0, w1, w2, ws);

  const float* Bp0 = ws;
  const float* Bp1 = ws + 224*224;
  const float* Bp2 = Bp1 + 192*192;
  const int nblk = (N + ROWS - 1) / ROWS;
  so2_fused<<<nblk, NTHREADS, 0, stream>>>(x, R, Jf, b0, Bp0, Bp1, Bp2, out, N);
}